// FOFEReader_77421080478417
// MI455X (gfx1250) — compile-verified
//
#include <hip/hip_runtime.h>

// ---- fixed problem dimensions (from reference) ----
#define T_    809
#define M_    16
#define B_    4
#define DD    304
#define E_    300
#define LQ    30
#define N_    12824          // sum over s of min(16, T-s)
#define CIN   1212           // 3*DD + E
#define H4    1024
#define H2    512
#define BN_EPS 1e-5f
#define ALPHA_F 0.9f

// LDS strides (floats). Even -> 8B-aligned float2 LDS loads; not mult of 64 -> bank spread.
#define SX   1214            // X tile stride   (16 x 1214 floats = 77,696 B)
#define SH1  1026            // h1 stride       (16 x 1026 fits in same buffer)
#define SH2  514             // h2 stride

typedef __attribute__((ext_vector_type(2))) float v2f;
typedef __attribute__((ext_vector_type(8))) float v8f;

// ---------------- scan kernel: F (forward) and R (reverse) alpha-scans ----------------
__global__ void fofe_scan_kernel(const float* __restrict__ doc,
                                 float* __restrict__ F, float* __restrict__ R) {
    int c = blockIdx.x * blockDim.x + threadIdx.x;
    if (c >= B_ * DD) return;
    int b = c / DD, d = c % DD;
    const float* base = doc + (size_t)b * T_ * DD + d;
    float* fb = F + (size_t)b * T_ * DD + d;
    float* rb = R + (size_t)b * T_ * DD + d;
    float acc = 0.0f;
    for (int t = 0; t < T_; ++t) {
        acc = ALPHA_F * acc + base[(size_t)t * DD];
        fb[(size_t)t * DD] = acc;
    }
    acc = 0.0f;
    for (int t = T_ - 1; t >= 0; --t) {
        acc = ALPHA_F * acc + base[(size_t)t * DD];
        rb[(size_t)t * DD] = acc;
    }
}

// ---------------- query FOFE: qf[b,e] = sum_t alpha^(LQ-1-t) * q[b,t,e] ----------------
__global__ void fofe_qf_kernel(const float* __restrict__ q, float* __restrict__ qf) {
    int idx = blockIdx.x * blockDim.x + threadIdx.x;
    if (idx >= B_ * E_) return;
    int b = idx / E_, e = idx % E_;
    float sum = 0.0f;
    double p = 1.0;
    for (int t = LQ - 1; t >= 0; --t) {
        sum += (float)p * q[((size_t)b * LQ + t) * E_ + e];
        p *= 0.9;
    }
    qf[idx] = sum;
}

// ---------------- fused gather + MLP (WMMA f32) ----------------
__global__ __launch_bounds__(256)
void fofe_mlp_kernel(const float* __restrict__ F, const float* __restrict__ R,
                     const float* __restrict__ qf,
                     const float* __restrict__ W1, const float* __restrict__ g1,
                     const float* __restrict__ b1, const float* __restrict__ m1,
                     const float* __restrict__ v1,
                     const float* __restrict__ W2, const float* __restrict__ g2,
                     const float* __restrict__ b2, const float* __restrict__ m2,
                     const float* __restrict__ v2,
                     const float* __restrict__ W3, float* __restrict__ out) {
    __shared__ __align__(16) float lds[16 * SX];   // X tile, then h1, then h2 (overlaid)
    __shared__ int   sArr[16], eArr[16], bArr[16];
    __shared__ float cArr[16];

    const int tid   = threadIdx.x;
    const int wave  = tid >> 5;
    const int lane  = tid & 31;
    const int lhalf = lane & 15;
    const int khi   = (lane < 16) ? 0 : 2;   // K-offset for A/B fragment halves
    const int r0    = blockIdx.x * 16;       // first global candidate-row of this block

    // per-row candidate metadata
    if (tid < 16) {
        int g = r0 + tid;
        int b = g / N_;
        int n = g % N_;
        int s, span;
        if (n < (T_ - M_ + 1) * M_) {
            s = n / M_; span = n % M_;
        } else {
            int rem = n - (T_ - M_ + 1) * M_;
            int cnt = M_ - 1;
            s = T_ - M_ + 1;
            while (rem >= cnt) { rem -= cnt; --cnt; ++s; }
            span = rem;
        }
        sArr[tid] = s; eArr[tid] = s + span; bArr[tid] = b;
        double c = 1.0;
        for (int i = 0; i <= span; ++i) c *= 0.9;   // coef = alpha^(span+1)
        cArr[tid] = (float)c;
    }
    __syncthreads();

    // ---- build 16 x 1212 input tile X in LDS ----
    for (int idx = tid; idx < 16 * CIN; idx += 256) {
        int lr = idx / CIN;
        int c  = idx % CIN;
        int b = bArr[lr], s = sArr[lr], e = eArr[lr];
        float val;
        if (c < DD) {                                    // lctx = Fp[s]
            val = (s == 0) ? 0.0f : F[((size_t)b * T_ + (s - 1)) * DD + c];
        } else if (c < 2 * DD) {                         // cand = F[e] - coef*Fp[s]
            int d = c - DD;
            float fp = (s == 0) ? 0.0f : F[((size_t)b * T_ + (s - 1)) * DD + d];
            val = F[((size_t)b * T_ + e) * DD + d] - cArr[lr] * fp;
        } else if (c < 3 * DD) {                         // rctx = Rp[e+1]
            int d = c - 2 * DD;
            val = (e + 1 >= T_) ? 0.0f : R[((size_t)b * T_ + (e + 1)) * DD + d];
        } else {                                         // broadcast query FOFE
            val = qf[b * E_ + (c - 3 * DD)];
        }
        lds[lr * SX + c] = val;
    }
    __syncthreads();

    // ---- GEMM1: h1 = relu(bn(X @ W1^T)), wave owns 128 of 1024 cols ----
    const v8f vzero = {};
    v8f acc[8];
#pragma unroll
    for (int j = 0; j < 8; ++j) acc[j] = vzero;

    const int jw = wave * 128;
    for (int k0 = 0; k0 < CIN; k0 += 4) {
        v2f a = *(const v2f*)&lds[lhalf * SX + k0 + khi];
#pragma unroll
        for (int jt = 0; jt < 8; ++jt) {
            int col = jw + jt * 16 + lhalf;
            v2f bf = *(const v2f*)&W1[(size_t)col * CIN + k0 + khi];
            acc[jt] = __builtin_amdgcn_wmma_f32_16x16x4_f32(
                false, a, false, bf, (short)0, acc[jt], false, false);
        }
    }
    __syncthreads();   // X tile no longer needed; overlay with h1

    {
        const int rbase = (lane < 16) ? 0 : 8;
#pragma unroll
        for (int jt = 0; jt < 8; ++jt) {
            int col = jw + jt * 16 + lhalf;
            float sc = g1[col] * rsqrtf(v1[col] + BN_EPS);
            float sh = b1[col] - m1[col] * sc;
#pragma unroll
            for (int rr = 0; rr < 8; ++rr) {
                float h = acc[jt][rr] * sc + sh;
                lds[(rbase + rr) * SH1 + col] = (h > 0.0f) ? h : 0.0f;
            }
        }
    }
    __syncthreads();

    // ---- GEMM2: h2 = relu(bn(h1 @ W2^T)), wave owns 64 of 512 cols ----
    v8f acc2[4];
#pragma unroll
    for (int j = 0; j < 4; ++j) acc2[j] = vzero;

    const int jw2 = wave * 64;
    for (int k0 = 0; k0 < H4; k0 += 4) {
        v2f a = *(const v2f*)&lds[lhalf * SH1 + k0 + khi];
#pragma unroll
        for (int jt = 0; jt < 4; ++jt) {
            int col = jw2 + jt * 16 + lhalf;
            v2f bf = *(const v2f*)&W2[(size_t)col * H4 + k0 + khi];
            acc2[jt] = __builtin_amdgcn_wmma_f32_16x16x4_f32(
                false, a, false, bf, (short)0, acc2[jt], false, false);
        }
    }
    __syncthreads();   // h1 no longer needed; overlay with h2

    {
        const int rbase = (lane < 16) ? 0 : 8;
#pragma unroll
        for (int jt = 0; jt < 4; ++jt) {
            int col = jw2 + jt * 16 + lhalf;
            float sc = g2[col] * rsqrtf(v2[col] + BN_EPS);
            float sh = b2[col] - m2[col] * sc;
#pragma unroll
            for (int rr = 0; rr < 8; ++rr) {
                float h = acc2[jt][rr] * sc + sh;
                lds[(rbase + rr) * SH2 + col] = (h > 0.0f) ? h : 0.0f;
            }
        }
    }
    __syncthreads();

    // ---- GEMM3: out = h2 @ W3^T  (16 rows x 2 cols) ----
    if (tid < 32) {
        int row = tid & 15, oc = tid >> 4;
        float sum = 0.0f;
        for (int k = 0; k < H2; ++k)
            sum += lds[row * SH2 + k] * W3[oc * H2 + k];
        out[(size_t)(r0 + row) * 2 + oc] = sum;
    }
}

extern "C" void kernel_launch(void* const* d_in, const int* in_sizes, int n_in,
                              void* d_out, int out_size, void* d_ws, size_t ws_size,
                              hipStream_t stream) {
    const float* doc = (const float*)d_in[0];
    const float* qry = (const float*)d_in[1];
    const float* W1  = (const float*)d_in[2];
    const float* g1  = (const float*)d_in[3];
    const float* b1  = (const float*)d_in[4];
    const float* m1  = (const float*)d_in[5];
    const float* v1  = (const float*)d_in[6];
    const float* W2  = (const float*)d_in[7];
    const float* g2  = (const float*)d_in[8];
    const float* b2  = (const float*)d_in[9];
    const float* m2  = (const float*)d_in[10];
    const float* v2  = (const float*)d_in[11];
    const float* W3  = (const float*)d_in[12];
    float* out = (float*)d_out;

    float* F  = (float*)d_ws;                       // B*T*DD floats
    float* R  = F + (size_t)B_ * T_ * DD;           // B*T*DD floats
    float* qf = R + (size_t)B_ * T_ * DD;           // B*E floats  (~7.9 MB total)

    fofe_scan_kernel<<<(B_ * DD + 255) / 256, 256, 0, stream>>>(doc, F, R);
    fofe_qf_kernel<<<(B_ * E_ + 255) / 256, 256, 0, stream>>>(qry, qf);

    const int rows = B_ * N_;          // 51296, exactly divisible by 16
    fofe_mlp_kernel<<<rows / 16, 256, 0, stream>>>(F, R, qf,
                                                   W1, g1, b1, m1, v1,
                                                   W2, g2, b2, m2, v2,
                                                   W3, out);
}